// YoloLoss_9285719294295
// MI455X (gfx1250) — compile-verified
//
#include <hip/hip_runtime.h>
#include <math.h>

// ---------------------------------------------------------------------------
// Problem constants (from reference setup_inputs)
// ---------------------------------------------------------------------------
#define NB 16
#define NA 9
#define NH 128
#define NW 128
#define NC 4
#define TT 50
#define CSTRIDE 10          // 6 + NUM_CLASSES channels per cell
#define TSTRIDE 17          // 13 + NUM_CLASSES per target row
#define NCELLS (NB*NA*NH*NW)   // 2,359,296

// anchors = ANCHORS_PX / SCALE
__constant__ float c_aw[NA] = {1.f, 2.f, 4.f, 8.f, 16.f, 2.f, 4.f, 8.f, 4.f};
__constant__ float c_ah[NA] = {1.f, 2.f, 4.f, 8.f, 16.f, 4.f, 2.f, 4.f, 8.f};

typedef float v2f __attribute__((ext_vector_type(2)));
typedef float v8f __attribute__((ext_vector_type(8)));

__device__ __forceinline__ float bce_logits(float x, float z) {
    // max(x,0) - x*z + log1p(exp(-|x|))
    return fmaxf(x, 0.f) - x * z + log1pf(expf(-fabsf(x)));
}

__device__ __forceinline__ float inv_tanh_ref(float y) {
    if (y <= -1.f) return -2.f;
    if (y >=  1.f) return  2.f;
    return 0.5f * logf((1.f + y) / (1.f - y));
}

// ---------------------------------------------------------------------------
// Kernel 1: full-grid reduction over pred_conf.
//   ws[blockIdx]        = sum of bce(conf, 0) for this block's cells
//   ws[512 + blockIdx]  = count of conf > 0
//
// Data movement: per-lane strided conf values are GATHERED into contiguous
// per-wave LDS buffers with global_load_async_to_lds_b32 (ASYNCcnt), double
// buffered: issue batch it+1, s_wait_asynccnt 2 releases batch it (async
// loads complete in order). s_wait_dscnt 0 before reissue keeps the async
// engine from overwriting a buffer still being read.
// Reduction: V_WMMA_F32_16X16X4_F32 with all-ones A folds 64 fp32 values per
// instruction into persistent accumulators in the matrix pipe, co-executing
// with the VALU bce math. Kernel is HBM-bound: ~94MB of lines -> ~4us.
// ---------------------------------------------------------------------------
#define K1_BLOCKS  512
#define K1_THREADS 256
#define USE_ASYNC_LDS 1

__global__ __launch_bounds__(K1_THREADS)
void conf_grid_reduce(const float* __restrict__ pred, float* __restrict__ ws) {
    const int gtid = blockIdx.x * K1_THREADS + threadIdx.x;
    constexpr int STRIDE = K1_BLOCKS * K1_THREADS * 2;   // cells per sweep = 262144
    constexpr int ITERS  = NCELLS / STRIDE;              // == 9
    static_assert(NCELLS % STRIDE == 0, "exact tiling: no tail loop");
    const int wid = threadIdx.x >> 5, lane = threadIdx.x & 31;

    float cnt = 0.f;

#if __has_builtin(__builtin_amdgcn_wmma_f32_16x16x4_f32)
    v2f aones; aones[0] = 1.f; aones[1] = 1.f;
    v8f acc = {0.f, 0.f, 0.f, 0.f, 0.f, 0.f, 0.f, 0.f};
#else
    float ssum = 0.f;
#endif

#if USE_ASYNC_LDS
    __shared__ float abuf[K1_THREADS / 32][2][64];
    // LDS byte addresses for this lane's slots (flat shared addr low 32 bits
    // == LDS byte offset per ISA 10.2 aperture mapping).
    unsigned ldsA[2], ldsB[2];
    ldsA[0] = (unsigned)(size_t)&abuf[wid][0][lane];
    ldsA[1] = (unsigned)(size_t)&abuf[wid][1][lane];
    ldsB[0] = (unsigned)(size_t)&abuf[wid][0][32 + lane];
    ldsB[1] = (unsigned)(size_t)&abuf[wid][1][32 + lane];

    auto issue = [&](int it, int bb) {
        const int idx = gtid * 2 + it * STRIDE;
        const unsigned long long g0 =
            (unsigned long long)(size_t)(pred + (size_t)idx * CSTRIDE);
        const unsigned long long g1 =
            (unsigned long long)(size_t)(pred + (size_t)(idx + 1) * CSTRIDE);
        asm volatile("global_load_async_to_lds_b32 %0, %1, off"
                     :: "v"(ldsA[bb]), "v"(g0) : "memory");
        asm volatile("global_load_async_to_lds_b32 %0, %1, off"
                     :: "v"(ldsB[bb]), "v"(g1) : "memory");
    };
    issue(0, 0);
#endif

    // Uniform trip count, uniform branches only: EXEC stays all-ones for WMMA.
    for (int it = 0; it < ITERS; ++it) {
        float v0, v1;
#if USE_ASYNC_LDS
        const int bb = it & 1;
        if (it + 1 < ITERS) {
            // prior reads of the other buffer must land before async overwrite
            asm volatile("s_wait_dscnt 0x0" ::: "memory");
            issue(it + 1, bb ^ 1);
            // oldest 2 async loads (current buffer) complete in order
            asm volatile("s_wait_asynccnt 0x2" ::: "memory");
        } else {
            asm volatile("s_wait_asynccnt 0x0" ::: "memory");
        }
        v0 = abuf[wid][bb][lane];
        v1 = abuf[wid][bb][32 + lane];
#else
        const int idx = gtid * 2 + it * STRIDE;
        v0 = pred[(size_t)idx * CSTRIDE];
        v1 = pred[(size_t)(idx + 1) * CSTRIDE];
        __builtin_prefetch(pred + (size_t)(idx + STRIDE) * CSTRIDE, 0, 1);
#endif
        cnt += (v0 > 0.f) ? 1.f : 0.f;
        cnt += (v1 > 0.f) ? 1.f : 0.f;
        const float b0 = bce_logits(v0, 0.f);
        const float b1 = bce_logits(v1, 0.f);
#if __has_builtin(__builtin_amdgcn_wmma_f32_16x16x4_f32)
        v2f bv; bv[0] = b0; bv[1] = b1;
        // D[m][n] = sum_k 1*B[k][n] + C[m][n]: 64 values folded per instruction.
        acc = __builtin_amdgcn_wmma_f32_16x16x4_f32(
            false, aones, false, bv, (short)0, acc, false, false);
#else
        ssum += b0 + b1;
#endif
    }

#if __has_builtin(__builtin_amdgcn_wmma_f32_16x16x4_f32)
    float lane_val = acc[0];   // VGPR0: lanes 0-15 = row M=0, lanes 16-31 = row M=8 (dup)
    const float factor = 0.5f; // rows identical -> 32-lane sum double-counts
#else
    float lane_val = ssum;
    const float factor = 1.0f;
#endif
    // wave32 butterfly reduction
    for (int m = 16; m > 0; m >>= 1) {
        lane_val += __shfl_xor(lane_val, m, 32);
        cnt      += __shfl_xor(cnt,      m, 32);
    }
    const float wave_sum = lane_val * factor;

    __shared__ float s_sum[K1_THREADS / 32];
    __shared__ float s_cnt[K1_THREADS / 32];
    if (lane == 0) { s_sum[wid] = wave_sum; s_cnt[wid] = cnt; }
    __syncthreads();
    if (threadIdx.x == 0) {
        float a = 0.f, b = 0.f;
        for (int k = 0; k < K1_THREADS / 32; ++k) { a += s_sum[k]; b += s_cnt[k]; }
        ws[blockIdx.x]       = a;   // deterministic per-block partial
        ws[512 + blockIdx.x] = b;
    }
}

// ---------------------------------------------------------------------------
// Kernel 2: per-batch sequential replay of the target scan (preserves the
// last-writer-wins and conf_mask 1->0->1->0 toggling semantics), touched
// cells tracked in an LDS hash; then block-parallel corrections.
// Output per batch b at ws[1024 + b*12 + k]:
//   0 corr_conf   (replace base bce terms with (cm-mask)*bce(pc,tconf))
//   1 corr_cmf    (replace base +1 weight with (cm-mask))
//   2 mask_count  3 sum bce(pc,1) over mask cells
//   4..7 sum (x-tx)^2, (y-ty)^2, (w-tw)^2, (h-th)^2 over mask cells
//   8 sum ce over mask cells   9 nGT   10 nCorrect
// ---------------------------------------------------------------------------
#define HSZ 1024
#define K2_THREADS 64

__global__ __launch_bounds__(K2_THREADS)
void target_phase(const float* __restrict__ pred, const float* __restrict__ target,
                  const int* __restrict__ tsizes, float* __restrict__ ws) {
    const int b = blockIdx.x;

    __shared__ int           h_key[HSZ];
    __shared__ float         h_tx[HSZ], h_ty[HSZ], h_tw[HSZ], h_th[HSZ];
    __shared__ unsigned char h_cm[HSZ], h_mask[HSZ], h_lab[HSZ];
    __shared__ int           s_list[512];
    __shared__ int           s_n;
    __shared__ float         s_scalar[2];              // nGT, nCorrect
    __shared__ float         s_part[K2_THREADS * 12];

    for (int k = threadIdx.x; k < HSZ; k += blockDim.x) h_key[k] = -1;
    if (threadIdx.x == 0) s_n = 0;
    __syncthreads();

    if (threadIdx.x == 0) {
        auto findins = [&](int a, int j, int i) -> int {
            const int key = (a << 14) | (j << 7) | i;
            int slot = (int)((((unsigned)key * 2654435761u) >> 22) & (HSZ - 1));
            for (;;) {
                int k = h_key[slot];
                if (k == key) return slot;
                if (k == -1) {
                    h_key[slot] = key;
                    h_cm[slot] = 1; h_mask[slot] = 0; h_lab[slot] = 0;
                    h_tx[slot] = 0.f; h_ty[slot] = 0.f; h_tw[slot] = 0.f; h_th[slot] = 0.f;
                    s_list[s_n++] = slot;
                    return slot;
                }
                slot = (slot + 1) & (HSZ - 1);
            }
        };

        int nGT = 0, nCorrect = 0;
        const int ts = tsizes[b];
        for (int t = 0; t < TT; ++t) {
            const float* row = target + ((size_t)b * TT + t) * TSTRIDE;
            const float gx = row[0] * (1.f / 16.f);
            const float gy = row[1] * (1.f / 16.f);
            const float gh = row[3] * (1.f / 16.f);
            const float gw = row[4] * (1.f / 16.f);
            const bool valid = (t < ts) && (gw != 0.f) && (gh != 0.f);
            if (!valid) continue;

            int gi = (int)gx; gi = min(max(gi, 0), NW - 1);
            int gj = (int)gy; gj = min(max(gj, 0), NH - 1);

            float ious[NA];
            float best = -1.f; int bestn = 0;
            for (int a = 0; a < NA; ++a) {
                const float aw = c_aw[a], ah = c_ah[a];
                const float inter = (fminf(gw, aw) + 1.f) * (fminf(gh, ah) + 1.f);
                const float u = (gw + 1.f) * (gh + 1.f) + (aw + 1.f) * (ah + 1.f)
                                - inter + 1e-16f;
                const float iou = inter / u;
                ious[a] = iou;
                if (iou > best) { best = iou; bestn = a; }   // first-max semantics
            }
            for (int a = 0; a < NA; ++a)
                if (ious[a] > 0.5f) h_cm[findins(a, gj, gi)] = 0;   // ignore-zeroing

            const int slot = findins(bestn, gj, gi);
            h_cm[slot] = 1; h_mask[slot] = 1;
            h_tx[slot] = inv_tanh_ref(gx - ((float)gi + 0.5f));
            h_ty[slot] = inv_tanh_ref(gy - ((float)gj + 0.5f));
            h_tw[slot] = logf(gw / c_aw[bestn] + 1e-16f);
            h_th[slot] = logf(gh / c_ah[bestn] + 1e-16f);
            int tlab = 0; float tb = row[13];
            for (int c = 1; c < NC; ++c)
                if (row[13 + c] > tb) { tb = row[13 + c]; tlab = c; }
            h_lab[slot] = (unsigned char)tlab;
            ++nGT;

            // recall / precision bookkeeping (pure values)
            const size_t base = ((((size_t)b * NA + bestn) * NH + gj) * NW + gi) * CSTRIDE;
            const float pc = pred[base + 0];
            const float px = tanhf(pred[base + 1]) + 0.5f + (float)gi;
            const float py = tanhf(pred[base + 2]) + 0.5f + (float)gj;
            const float ph = expf(pred[base + 4]) * c_ah[bestn];
            const float pw = expf(pred[base + 5]) * c_aw[bestn];
            const float b1x1 = gx - gw, b1x2 = gx + gw, b1y1 = gy - gh, b1y2 = gy + gh;
            const float b2x1 = px - pw, b2x2 = px + pw, b2y1 = py - ph, b2y2 = py + ph;
            const float iw = fmaxf(fminf(b1x2, b2x2) - fmaxf(b1x1, b2x1) + 1.f, 0.f);
            const float ih = fmaxf(fminf(b1y2, b2y2) - fmaxf(b1y1, b2y1) + 1.f, 0.f);
            const float inter = iw * ih;
            const float a1 = (b1x2 - b1x1 + 1.f) * (b1y2 - b1y1 + 1.f);
            const float a2 = (b2x2 - b2x1 + 1.f) * (b2y2 - b2y1 + 1.f);
            const float iou = inter / (a1 + a2 - inter + 1e-16f);
            int plab = 0; float pb = pred[base + 6];
            for (int c = 1; c < NC; ++c) {
                const float v = pred[base + 6 + c];
                if (v > pb) { pb = v; plab = c; }
            }
            if (iou > 0.5f && plab == tlab && pc > 0.f) ++nCorrect;
        }
        s_scalar[0] = (float)nGT;
        s_scalar[1] = (float)nCorrect;
    }
    __syncthreads();

    // Phase B: corrections over touched cells, parallel across the block.
    float p[9];
    for (int k = 0; k < 9; ++k) p[k] = 0.f;
    const int n = s_n;
    for (int e = threadIdx.x; e < n; e += blockDim.x) {
        const int slot = s_list[e];
        const int key  = h_key[slot];
        const int a = key >> 14, j = (key >> 7) & 127, i = key & 127;
        const size_t base = ((((size_t)b * NA + a) * NH + j) * NW + i) * CSTRIDE;
        const float pc   = pred[base];
        const float bce0 = bce_logits(pc, 0.f);
        const float cm = (float)h_cm[slot];
        const float mk = (float)h_mask[slot];            // tconf == mask
        const float bce_t = (mk > 0.f) ? bce_logits(pc, 1.f) : bce0;
        p[0] += (cm - mk) * bce_t - bce0;
        p[1] += (cm - mk) - 1.f;
        if (mk > 0.f) {
            p[2] += 1.f;
            p[3] += bce_t;
            const float x  = pred[base + 1], y = pred[base + 2];
            const float hh = pred[base + 4], wv = pred[base + 5];
            const float dx = x  - h_tx[slot]; p[4] += dx * dx;
            const float dy = y  - h_ty[slot]; p[5] += dy * dy;
            const float dw = wv - h_tw[slot]; p[6] += dw * dw;
            const float dh = hh - h_th[slot]; p[7] += dh * dh;
            const float c0 = pred[base + 6], c1 = pred[base + 7];
            const float c2 = pred[base + 8], c3 = pred[base + 9];
            const float m = fmaxf(fmaxf(c0, c1), fmaxf(c2, c3));
            const float lse = m + logf(expf(c0 - m) + expf(c1 - m) +
                                       expf(c2 - m) + expf(c3 - m));
            const int lab = (int)h_lab[slot];
            const float sel = (lab == 0) ? c0 : (lab == 1) ? c1 : (lab == 2) ? c2 : c3;
            p[8] += lse - sel;
        }
    }
    for (int k = 0; k < 9; ++k) s_part[threadIdx.x * 12 + k] = p[k];
    __syncthreads();
    if (threadIdx.x == 0) {
        float acc[9];
        for (int k = 0; k < 9; ++k) acc[k] = 0.f;
        for (int td = 0; td < K2_THREADS; ++td)        // fixed order: deterministic
            for (int k = 0; k < 9; ++k) acc[k] += s_part[td * 12 + k];
        float* o = ws + 1024 + b * 12;
        for (int k = 0; k < 9; ++k) o[k] = acc[k];
        o[9]  = s_scalar[0];
        o[10] = s_scalar[1];
    }
}

// ---------------------------------------------------------------------------
// Kernel 3: fold partials (fixed order -> deterministic) into the 6 outputs.
// ---------------------------------------------------------------------------
__global__ void finalize(const float* __restrict__ ws, float* __restrict__ out) {
    if (threadIdx.x != 0 || blockIdx.x != 0) return;
    float S0 = 0.f, nProp = 0.f;
    for (int k = 0; k < K1_BLOCKS; ++k) { S0 += ws[k]; nProp += ws[512 + k]; }
    float cc = 0.f, ccmf = 0.f, mcnt = 0.f, scm = 0.f;
    float sx = 0.f, sy = 0.f, sw = 0.f, sh = 0.f, sce = 0.f, nGT = 0.f, nCor = 0.f;
    for (int bb = 0; bb < NB; ++bb) {
        const float* o = ws + 1024 + bb * 12;
        cc += o[0]; ccmf += o[1]; mcnt += o[2]; scm += o[3];
        sx += o[4]; sy += o[5]; sw += o[6]; sh += o[7];
        sce += o[8]; nGT += o[9]; nCor += o[10];
    }
    const float sum_cmf     = (float)NCELLS + ccmf;
    const float sum_conf_un = S0 + cc;
    const float d_cmf = fmaxf(sum_cmf, 1.f);
    const float d_m   = fmaxf(mcnt, 1.f);
    const float loss_conf = 1.25f * (sum_conf_un / d_cmf) + (scm / d_m);
    const float lx = sx / d_m, ly = sy / d_m, lw = sw / d_m, lh = sh / d_m;
    const float lcls = (sce / d_m) / (float)NB;
    const float coord = lx + ly + lw + lh;
    const float loss = coord + loss_conf + lcls;
    const float recall    = (nGT  > 0.f) ? (nCor / fmaxf(nGT,  1.f)) : 1.f;
    const float precision = (nProp > 0.f) ? (nCor / fmaxf(nProp, 1.f)) : 1.f;
    out[0] = loss; out[1] = coord; out[2] = loss_conf;
    out[3] = lcls; out[4] = recall; out[5] = precision;
}

// ---------------------------------------------------------------------------
extern "C" void kernel_launch(void* const* d_in, const int* in_sizes, int n_in,
                              void* d_out, int out_size, void* d_ws, size_t ws_size,
                              hipStream_t stream) {
    (void)in_sizes; (void)n_in; (void)out_size; (void)ws_size;
    const float* pred   = (const float*)d_in[0];
    const float* target = (const float*)d_in[1];
    const int*   tsz    = (const int*)d_in[2];
    float* ws  = (float*)d_ws;    // [0..511] bce partials, [512..1023] proposal counts,
                                  // [1024..1215] 16 batches x 12 correction partials
    float* out = (float*)d_out;   // 6 fp32 scalars

    conf_grid_reduce<<<K1_BLOCKS, K1_THREADS, 0, stream>>>(pred, ws);
    target_phase   <<<NB,        K2_THREADS, 0, stream>>>(pred, target, tsz, ws);
    finalize       <<<1,         32,         0, stream>>>(ws, out);
}